// MyGIN_51221779972720
// MI455X (gfx1250) — compile-verified
//
#include <hip/hip_runtime.h>
#include <hip/hip_bf16.h>

typedef __attribute__((ext_vector_type(2))) float v2f;
typedef __attribute__((ext_vector_type(8))) float v8f;

#define N_NODES 100000
#define N_EDGES 1600000
#define N_GRAPHS 512
#define WST 72   // row-major W stride in LDS floats (conflict-free for B-frag reads)

__device__ __forceinline__ v8f wmma_f32_k4(v2f a, v2f b, v8f c) {
    // V_WMMA_F32_16X16X4_F32 : D = A(16x4) * B(4x16) + C, full fp32
    return __builtin_amdgcn_wmma_f32_16x16x4_f32(false, a, false, b, (short)0, c, false, false);
}

__global__ void zero_kernel(float* __restrict__ p, long n) {
    long i = (long)blockIdx.x * blockDim.x + threadIdx.x;
    long stride = (long)gridDim.x * blockDim.x;
    for (; i < n; i += stride) p[i] = 0.0f;
}

// Edge-parallel gather + scatter-add. Each thread: one edge, 4 consecutive features.
__global__ void aggregate_kernel(const float* __restrict__ x, int xstride, int xoff,
                                 const int* __restrict__ src, const int* __restrict__ dst,
                                 float* __restrict__ agg, int F, int chunk_shift, long total) {
    long tid = (long)blockIdx.x * blockDim.x + threadIdx.x;
    if (tid >= total) return;
    long e = tid >> chunk_shift;
    int c = (int)(tid & ((1 << chunk_shift) - 1)) << 2;
    int s = src[e];
    int d = dst[e];
    const float4 v = *(const float4*)(x + (long)s * xstride + xoff + c);
    float* ap = agg + (long)d * F + c;
    atomicAdd(ap + 0, v.x);
    atomicAdd(ap + 1, v.y);
    atomicAdd(ap + 2, v.z);
    atomicAdd(ap + 3, v.w);
}

// One block = 256 threads = 8 waves; block computes a 128-row x 64-col output tile.
// mode 1: in = (1+eps)*x + agg            -> out = in@W + b        (pre-BN, no relu)
// mode 2: in = relu(bn(t))                -> out = relu(in@W + b)  (layer output)
__global__ void gin_gemm_kernel(
    int mode,
    const float* __restrict__ xbase, int xstride, int xoff,
    const float* __restrict__ agg,
    const float* __restrict__ epsv, int layer,
    const float* __restrict__ tin,
    const float* __restrict__ stats, float invn,
    const float* __restrict__ gamma, const float* __restrict__ beta,
    const float* __restrict__ W, const float* __restrict__ bias,
    float* __restrict__ out, int outstride, int outoff, int dorelu,
    int K, int kshift, int n)
{
    extern __shared__ float smem[];
    const int ZS = K + 4;                 // z stride ≡ 4 (mod 64): conflict-free A-frag reads
    float* z  = smem;                     // [128][ZS] staged/normalized input tile
    float* wl = smem + 128 * ZS;          // [K][WST]  W row-major
    float* mu = wl + K * WST;             // [64]
    float* rs = mu + 64;                  // [64]
    const unsigned wbase_bytes = (unsigned)(128 * ZS) * 4u;

    const int tid = threadIdx.x;
    const long r0 = (long)blockIdx.x * 128;

    if (mode == 2) {
        if (tid < 64) {
            float s = stats[tid] * invn;
            float q = stats[64 + tid] * invn;
            mu[tid] = s;
            rs[tid] = rsqrtf(q - s * s + 1e-5f);
        }
        __syncthreads();
    }

    // Async-copy W (global -> LDS, row-major, padded stride) bypassing VGPRs.
    // Each issue moves 16 bytes per lane; ASYNCcnt tracks completion.
    {
        const int chunks = K << 4;               // (K*64)/4 four-float chunks
        for (int i = tid; i < chunks; i += blockDim.x) {
            int k  = i >> 4;                     // 16 chunks per 64-float row
            int c4 = (i & 15) << 2;
            unsigned ldsb = wbase_bytes + (unsigned)(k * WST + c4) * 4u;
            const float* gp = W + ((long)i << 2);
            asm volatile("global_load_async_to_lds_b128 %0, %1, off"
                         :: "v"(ldsb), "v"(gp)
                         : "memory");
        }
    }

    // Stage input tile into LDS (zero-pad rows >= n so WMMA math is safe);
    // overlaps with the in-flight async W copy.
    if (mode == 1) {
        const float ep = 1.0f + epsv[layer];
        for (int i = tid; i < 128 * K; i += blockDim.x) {
            int r = i >> kshift, c = i & (K - 1);
            long gr = r0 + r;
            float v = 0.0f;
            if (gr < n) v = ep * xbase[gr * (long)xstride + xoff + c] + agg[gr * (long)K + c];
            z[r * ZS + c] = v;
        }
    } else {
        for (int i = tid; i < 128 * K; i += blockDim.x) {
            int r = i >> kshift, c = i & (K - 1);
            long gr = r0 + r;
            float v = 0.0f;
            if (gr < n) {
                float v0 = tin[gr * 64 + c];
                v = fmaxf((v0 - mu[c]) * rs[c] * gamma[c] + beta[c], 0.0f);
            }
            z[r * ZS + c] = v;
        }
    }
    asm volatile("s_wait_asynccnt 0x0" ::: "memory");  // our wave's async W writes done
    __syncthreads();                                   // publish LDS across waves

    const int wv   = tid >> 5;            // wave id 0..7 -> 16-row strip
    const int lane = tid & 31;
    const int ml   = (wv << 4) + (lane & 15);   // A row (lanes 0-15 & 16-31 share M)
    const int kh   = (lane >> 4) << 1;          // lanes 16-31 hold K=+2,+3
    const int nl   = lane & 15;

    v8f acc[4] = {};
    for (int k = 0; k < K; k += 4) {
        const int kb = k + kh;
        v2f a;
        a.x = z[ml * ZS + kb];
        a.y = z[ml * ZS + kb + 1];
#pragma unroll
        for (int nt = 0; nt < 4; nt++) {
            const int nn = nt * 16 + nl;
            v2f b;
            b.x = wl[kb * WST + nn];
            b.y = wl[(kb + 1) * WST + nn];
            acc[nt] = wmma_f32_k4(a, b, acc[nt]);
        }
    }

    // Epilogue: C/D layout — lane gives N; VGPR v gives M = v + 8*(lane>=16)
    const int hh = lane >> 4;
#pragma unroll
    for (int nt = 0; nt < 4; nt++) {
        const int nn = nt * 16 + nl;
        const float bs = bias[nn];
#pragma unroll
        for (int v = 0; v < 8; v++) {
            long gr = r0 + (wv << 4) + (hh << 3) + v;
            if (gr < n) {
                float val = acc[nt][v] + bs;
                if (dorelu) val = fmaxf(val, 0.0f);
                out[gr * (long)outstride + outoff + nn] = val;
            }
        }
    }
}

// Column sum / sum-of-squares over t[n][64] -> stats[0..63]=sum, stats[64..127]=sumsq
__global__ void stats_kernel(const float* __restrict__ t, float* __restrict__ stats, int n) {
    __shared__ float ssum[64], ssq[64];
    int tid = threadIdx.x;
    if (tid < 64) { ssum[tid] = 0.0f; ssq[tid] = 0.0f; }
    __syncthreads();
    int c = tid & 63;
    long r = ((long)blockIdx.x * blockDim.x + tid) >> 6;
    long stride = ((long)gridDim.x * blockDim.x) >> 6;
    float s = 0.0f, q = 0.0f;
    for (; r < n; r += stride) {
        float v = t[r * 64 + c];
        s += v; q += v * v;
    }
    atomicAdd(&ssum[c], s);
    atomicAdd(&ssq[c], q);
    __syncthreads();
    if (tid < 64) {
        atomicAdd(&stats[tid], ssum[tid]);
        atomicAdd(&stats[64 + tid], ssq[tid]);
    }
}

__global__ void pool_kernel(const float* __restrict__ hcat, const int* __restrict__ batch,
                            float* __restrict__ pooled, int n) {
    long tid = (long)blockIdx.x * blockDim.x + threadIdx.x;
    long total = (long)n * 64;
    if (tid >= total) return;
    int node = (int)(tid >> 6);
    int c = (int)(tid & 63) << 2;
    int g = batch[node];
    float4 v = *(const float4*)(hcat + (long)node * 256 + c);
    float* p = pooled + (long)g * 256 + c;
    atomicAdd(p + 0, v.x);
    atomicAdd(p + 1, v.y);
    atomicAdd(p + 2, v.z);
    atomicAdd(p + 3, v.w);
}

__global__ void head1_kernel(const float* __restrict__ pooled, const float* __restrict__ W,
                             const float* __restrict__ b, float* __restrict__ hid) {
    int idx = blockIdx.x * blockDim.x + threadIdx.x;
    if (idx >= N_GRAPHS * 64) return;
    int c = idx & 63, r = idx >> 6;
    float acc = b[c];
    const float* pr = pooled + (long)r * 256;
    for (int k = 0; k < 256; k++) acc += pr[k] * W[k * 64 + c];
    hid[idx] = acc;
}

__global__ void head2_kernel(const float* __restrict__ hid, float* __restrict__ murs) {
    int c = threadIdx.x;               // 64 threads
    float s = 0.0f, q = 0.0f;
    for (int r = 0; r < N_GRAPHS; r++) {
        float v = hid[r * 64 + c];
        s += v; q += v * v;
    }
    float m = s * (1.0f / N_GRAPHS);
    float var = q * (1.0f / N_GRAPHS) - m * m;
    murs[c] = m;
    murs[64 + c] = rsqrtf(var + 1e-5f);
}

__global__ void head3_kernel(const float* __restrict__ hid, const float* __restrict__ murs,
                             const float* __restrict__ g, const float* __restrict__ be,
                             const float* __restrict__ W2, const float* __restrict__ b2,
                             float* __restrict__ out) {
    int r = blockIdx.x * blockDim.x + threadIdx.x;
    if (r >= N_GRAPHS) return;
    float h[64];
#pragma unroll
    for (int c = 0; c < 64; c++) {
        float v = (hid[r * 64 + c] - murs[c]) * murs[64 + c] * g[c] + be[c];
        h[c] = fmaxf(v, 0.0f);
    }
    float zz[10];
    float mx = -1e30f;
#pragma unroll
    for (int j = 0; j < 10; j++) {
        float a = b2[j];
        for (int c = 0; c < 64; c++) a += h[c] * W2[c * 10 + j];
        zz[j] = a;
        mx = fmaxf(mx, a);
    }
    float se = 0.0f;
#pragma unroll
    for (int j = 0; j < 10; j++) se += __expf(zz[j] - mx);
    float lse = mx + __logf(se);
#pragma unroll
    for (int j = 0; j < 10; j++) out[r * 10 + j] = zz[j] - lse;
}

extern "C" void kernel_launch(void* const* d_in, const int* in_sizes, int n_in,
                              void* d_out, int out_size, void* d_ws, size_t ws_size,
                              hipStream_t stream) {
    const float* x      = (const float*)d_in[0];
    const int*   ei     = (const int*)d_in[1];
    const int*   batch  = (const int*)d_in[2];
    const float* eps    = (const float*)d_in[3];
    const float* W1_0   = (const float*)d_in[4];
    const float* b1_0   = (const float*)d_in[5];
    const float* g_0    = (const float*)d_in[6];
    const float* be_0   = (const float*)d_in[7];
    const float* W2_0   = (const float*)d_in[8];
    const float* b2_0   = (const float*)d_in[9];
    const float* W1_r   = (const float*)d_in[10];
    const float* b1_r   = (const float*)d_in[11];
    const float* g_r    = (const float*)d_in[12];
    const float* be_r   = (const float*)d_in[13];
    const float* W2_r   = (const float*)d_in[14];
    const float* b2_r   = (const float*)d_in[15];
    const float* lin1_W = (const float*)d_in[16];
    const float* lin1_b = (const float*)d_in[17];
    const float* bn_g   = (const float*)d_in[18];
    const float* bn_b   = (const float*)d_in[19];
    const float* lin2_W = (const float*)d_in[20];
    const float* lin2_b = (const float*)d_in[21];

    const int* src = ei;
    const int* dst = ei + N_EDGES;

    // workspace layout (floats)
    float* ws     = (float*)d_ws;
    float* agg    = ws;                         // 100000*128
    float* t      = agg + (long)N_NODES * 128;  // 100000*64
    float* hcat   = t + (long)N_NODES * 64;     // 100000*256
    float* stats  = hcat + (long)N_NODES * 256; // 128
    float* pooled = stats + 128;                // 512*256
    float* hid    = pooled + N_GRAPHS * 256;    // 512*64
    float* murs   = hid + N_GRAPHS * 64;        // 128

    const int N = N_NODES;
    const int gemm_blocks = (N + 127) / 128;

    for (int l = 0; l < 4; l++) {
        const int K      = (l == 0) ? 128 : 64;
        const int kshift = (l == 0) ? 7 : 6;
        const float* xb  = (l == 0) ? x : hcat;
        const int xstride = (l == 0) ? 128 : 256;
        const int xoff    = (l == 0) ? 0 : (l - 1) * 64;
        const float* W1 = (l == 0) ? W1_0 : W1_r + (long)(l - 1) * 64 * 64;
        const float* b1 = (l == 0) ? b1_0 : b1_r + (l - 1) * 64;
        const float* gg = (l == 0) ? g_0  : g_r  + (l - 1) * 64;
        const float* bb = (l == 0) ? be_0 : be_r + (l - 1) * 64;
        const float* W2 = (l == 0) ? W2_0 : W2_r + (long)(l - 1) * 64 * 64;
        const float* b2 = (l == 0) ? b2_0 : b2_r + (l - 1) * 64;

        // 1) agg = segment_sum(x[src] -> dst)
        zero_kernel<<<2048, 256, 0, stream>>>(agg, (long)N * K);
        const int chunk_shift = (l == 0) ? 5 : 4;
        const long tot = (long)N_EDGES << chunk_shift;
        aggregate_kernel<<<(int)((tot + 255) / 256), 256, 0, stream>>>(
            xb, xstride, xoff, src, dst, agg, K, chunk_shift, tot);

        // 2) t = ((1+eps)*x + agg) @ W1 + b1   (WMMA fp32)
        size_t sh1 = (size_t)(128 * (K + 4) + K * WST + 128) * sizeof(float);
        gin_gemm_kernel<<<gemm_blocks, 256, sh1, stream>>>(
            1, xb, xstride, xoff, agg, eps, l,
            nullptr, nullptr, 0.0f, nullptr, nullptr,
            W1, b1, t, 64, 0, 0, K, kshift, N);

        // 3) batch-norm statistics of t
        zero_kernel<<<1, 128, 0, stream>>>(stats, 128);
        stats_kernel<<<1024, 256, 0, stream>>>(t, stats, N);

        // 4) hcat[:, l*64:] = relu(relu(bn(t)) @ W2 + b2)   (WMMA fp32)
        size_t sh2 = (size_t)(128 * 68 + 64 * WST + 128) * sizeof(float);
        gin_gemm_kernel<<<gemm_blocks, 256, sh2, stream>>>(
            2, nullptr, 0, 0, nullptr, nullptr, 0,
            t, stats, 1.0f / (float)N, gg, bb,
            W2, b2, hcat, 256, l * 64, 1, 64, 6, N);
    }

    // graph pooling
    zero_kernel<<<512, 256, 0, stream>>>(pooled, (long)N_GRAPHS * 256);
    {
        long tot = (long)N * 64;
        pool_kernel<<<(int)((tot + 255) / 256), 256, 0, stream>>>(hcat, batch, pooled, N);
    }

    // head
    head1_kernel<<<(N_GRAPHS * 64 + 255) / 256, 256, 0, stream>>>(pooled, lin1_W, lin1_b, hid);
    head2_kernel<<<1, 64, 0, stream>>>(hid, murs);
    head3_kernel<<<2, 256, 0, stream>>>(hid, murs, bn_g, bn_b, lin2_W, lin2_b, (float*)d_out);
}